// EPFGNN_94489280546
// MI455X (gfx1250) — compile-verified
//
#include <hip/hip_runtime.h>
#include <math.h>

#define N_NODES 20000
#define E_UND   320000
#define ED      (2*E_UND)
#define F_INN   128
#define HIDN    128
#define NC      8
#define BP_ITERS 100

typedef __attribute__((ext_vector_type(2))) float v2f;
typedef __attribute__((ext_vector_type(8))) float v8f;

// ---------------------------------------------------------------- CSR build
__global__ void k_zero_int(int* __restrict__ p, int n) {
    int i = blockIdx.x * blockDim.x + threadIdx.x;
    if (i < n) p[i] = 0;
}

// in-degree over directed edges: cnt[v] = #occurrences of v in d0 plus in s0
__global__ void k_indeg(const int* __restrict__ s0, const int* __restrict__ d0,
                        int* __restrict__ cnt, int e) {
    int i = blockIdx.x * blockDim.x + threadIdx.x;
    if (i < e) {
        atomicAdd(&cnt[d0[i]], 1);   // directed edge i     : dst = d0[i]
        atomicAdd(&cnt[s0[i]], 1);   // directed edge i + E : dst = s0[i]
    }
}

// single-block exclusive prefix scan of cnt[0..N) -> ptr, cursor; ptr[N]=total
__global__ void k_scan(const int* __restrict__ cnt, int* __restrict__ ptr,
                       int* __restrict__ cursor) {
    __shared__ int s[1024];
    const int CH = (N_NODES + 1023) / 1024;   // 20
    int t = threadIdx.x;
    int base = t * CH;
    int local[CH];
    int sum = 0;
#pragma unroll
    for (int i = 0; i < CH; ++i) {
        int idx = base + i;
        int v = (idx < N_NODES) ? cnt[idx] : 0;
        local[i] = v;
        sum += v;
    }
    s[t] = sum;
    __syncthreads();
    for (int off = 1; off < 1024; off <<= 1) {
        int v = (t >= off) ? s[t - off] : 0;
        __syncthreads();
        s[t] += v;
        __syncthreads();
    }
    int run = s[t] - sum;  // exclusive base of this chunk
#pragma unroll
    for (int i = 0; i < CH; ++i) {
        int idx = base + i;
        if (idx < N_NODES) {
            ptr[idx] = run;
            cursor[idx] = run;
            run += local[i];
        }
    }
    if (t == 1023) ptr[N_NODES] = s[1023];
}

// place directed edge ids + source nodes into incoming lists
__global__ void k_place(const int* __restrict__ s0, const int* __restrict__ d0,
                        int* __restrict__ cursor, int* __restrict__ eid,
                        int* __restrict__ esrc) {
    int e = blockIdx.x * blockDim.x + threadIdx.x;
    if (e >= ED) return;
    int s, d;
    if (e < E_UND) { s = s0[e]; d = d0[e]; }
    else           { s = d0[e - E_UND]; d = s0[e - E_UND]; }
    int k = atomicAdd(&cursor[d], 1);
    eid[k] = e;
    esrc[k] = s;
}

// dinv[v] = rsqrt(indeg + self-loop)
__global__ void k_dinv(const int* __restrict__ cnt, float* __restrict__ dinv, int n) {
    int i = blockIdx.x * blockDim.x + threadIdx.x;
    if (i < n) dinv[i] = rsqrtf((float)(cnt[i] + 1));
}

// ---------------------------------------------------------------- GEMM 1: H1 = x @ W1
// one wave per 16-row strip: 8 independent accumulators, A fragment reused 8x.
__global__ void k_gemm1(const float* __restrict__ X, const float* __restrict__ W,
                        float* __restrict__ O) {
    int row0 = blockIdx.x * 16;       // 1250 strips
    int lane = threadIdx.x & 31;
    int half = lane >> 4;             // K pair select
    int lm   = lane & 15;
    v8f acc[8];
#pragma unroll
    for (int t = 0; t < 8; ++t) acc[t] = (v8f){0.f,0.f,0.f,0.f,0.f,0.f,0.f,0.f};
    const float* arow = X + (long long)(row0 + lm) * F_INN;
    for (int k0 = 0; k0 < F_INN; k0 += 4) {
        int ka = k0 + 2 * half;
        v2f a; a.x = arow[ka]; a.y = arow[ka + 1];
        const float* w0 = W + ka * HIDN + lm;
        const float* w1 = W + (ka + 1) * HIDN + lm;
#pragma unroll
        for (int t = 0; t < 8; ++t) {
            v2f b; b.x = w0[t * 16]; b.y = w1[t * 16];
            acc[t] = __builtin_amdgcn_wmma_f32_16x16x4_f32(false, a, false, b,
                                                           (short)0, acc[t], false, false);
        }
    }
#pragma unroll
    for (int t = 0; t < 8; ++t)
#pragma unroll
        for (int i = 0; i < 8; ++i)
            O[(long long)(row0 + i + 8 * half) * HIDN + t * 16 + lm] = acc[t][i];
}

// ---------------------------------------------------------------- GEMM 2: G2 = h @ W2
__global__ void k_gemm2(const float* __restrict__ H, const float* __restrict__ W,
                        float* __restrict__ O) {
    int row0 = blockIdx.x * 16;       // 1250 tiles
    int lane = threadIdx.x & 31;
    int half = lane >> 4, lm = lane & 15;
    v8f acc = {0.f,0.f,0.f,0.f,0.f,0.f,0.f,0.f};
    const float* arow = H + (long long)(row0 + lm) * HIDN;
    for (int k0 = 0; k0 < HIDN; k0 += 4) {
        int ka = k0 + 2 * half;
        v2f a; a.x = arow[ka]; a.y = arow[ka + 1];
        v2f b;
        b.x = (lm < NC) ? W[ka * NC + lm]       : 0.0f;
        b.y = (lm < NC) ? W[(ka + 1) * NC + lm] : 0.0f;
        acc = __builtin_amdgcn_wmma_f32_16x16x4_f32(false, a, false, b,
                                                    (short)0, acc, false, false);
    }
    if (lm < NC) {
#pragma unroll
        for (int i = 0; i < 8; ++i)
            O[(long long)(row0 + i + 8 * half) * NC + lm] = acc[i];
    }
}

// ---------------------------------------------------------------- conv1 gather (fused self+bias+relu)
// one thread per (node, 4 features)
__global__ void k_conv1(const float* __restrict__ H, const int* __restrict__ ptr,
                        const int* __restrict__ esrc, const float* __restrict__ dinv,
                        const float* __restrict__ b1, float* __restrict__ out) {
    int idx = blockIdx.x * blockDim.x + threadIdx.x;  // N*32
    if (idx >= N_NODES * 32) return;
    int v = idx >> 5;
    int f4 = (idx & 31) << 2;
    int k0 = ptr[v], k1 = ptr[v + 1];
    float dv = dinv[v];
    float ax = 0.f, ay = 0.f, az = 0.f, aw = 0.f;
    for (int k = k0; k < k1; ++k) {
        int s = esrc[k];
        float w = dinv[s] * dv;
        const float4 hv = *reinterpret_cast<const float4*>(H + (long long)s * F_INN + f4);
        ax += hv.x * w; ay += hv.y * w; az += hv.z * w; aw += hv.w * w;
    }
    const float4 hs = *reinterpret_cast<const float4*>(H + (long long)v * F_INN + f4);
    const float4 bb = *reinterpret_cast<const float4*>(b1 + f4);
    float ws = dv * dv;
    float4 r;
    r.x = fmaxf(ax + hs.x * ws + bb.x, 0.f);
    r.y = fmaxf(ay + hs.y * ws + bb.y, 0.f);
    r.z = fmaxf(az + hs.z * ws + bb.z, 0.f);
    r.w = fmaxf(aw + hs.w * ws + bb.w, 0.f);
    *reinterpret_cast<float4*>(out + (long long)v * F_INN + f4) = r;
}

// ---------------------------------------------------------------- conv2 gather (fused self+bias)
// one thread per (node, class)
__global__ void k_conv2(const float* __restrict__ G, const int* __restrict__ ptr,
                        const int* __restrict__ esrc, const float* __restrict__ dinv,
                        const float* __restrict__ b2, float* __restrict__ U) {
    int idx = blockIdx.x * blockDim.x + threadIdx.x;  // N*8
    if (idx >= N_NODES * NC) return;
    int v = idx >> 3, c = idx & 7;
    int k0 = ptr[v], k1 = ptr[v + 1];
    float dv = dinv[v];
    float acc = 0.f;
    for (int k = k0; k < k1; ++k) {
        int s = esrc[k];
        acc += G[s * NC + c] * dinv[s] * dv;
    }
    U[idx] = acc + G[v * NC + c] * dv * dv + b2[c];
}

// ---------------------------------------------------------------- node potentials (in place)
__global__ void k_phi_u(float* __restrict__ U, const int* __restrict__ target,
                        const unsigned char* __restrict__ mask) {
    int v = blockIdx.x * blockDim.x + threadIdx.x;
    if (v >= N_NODES) return;
    float e[NC];
#pragma unroll
    for (int c = 0; c < NC; ++c) e[c] = -U[v * NC + c];
    int t = target[v];
    float gt = e[t];
    if (mask[v]) {
#pragma unroll
        for (int c = 0; c < NC; ++c) e[c] = 1e5f;
        e[t] = gt;
    }
    float mn = e[0];
#pragma unroll
    for (int c = 1; c < NC; ++c) mn = fminf(mn, e[c]);
#pragma unroll
    for (int c = 0; c < NC; ++c)
        U[v * NC + c] = -fminf(e[c] - mn, 20.0f);
}

// phi_b = B_sym - max(B_sym)
__global__ void k_phi_b(const float* __restrict__ binary, float* __restrict__ phib) {
    __shared__ float s[64];
    __shared__ float mx;
    int i = threadIdx.x;  // 64 threads
    int r = i >> 3, c = i & 7;
    float bs = 0.5f * (binary[r * 8 + c] + binary[c * 8 + r]);
    s[i] = bs;
    __syncthreads();
    if (i == 0) {
        float m = s[0];
        for (int k = 1; k < 64; ++k) m = fmaxf(m, s[k]);
        mx = m;
    }
    __syncthreads();
    phib[i] = bs - mx;
}

__global__ void k_fill(float* __restrict__ p, float v, int n) {
    int i = blockIdx.x * blockDim.x + threadIdx.x;
    if (i < n) p[i] = v;
}

// ---------------------------------------------------------------- BP: gather incoming messages
// node_in[v,c] = sum over incoming directed edges of m[e,c]; no atomics, direct write
__global__ void k_gather_m(const float* __restrict__ m, const int* __restrict__ ptr,
                           const int* __restrict__ eid, float* __restrict__ node_in) {
    int idx = blockIdx.x * blockDim.x + threadIdx.x;  // N*8
    if (idx >= N_NODES * NC) return;
    int v = idx >> 3, c = idx & 7;
    int k0 = ptr[v], k1 = ptr[v + 1];
    float acc = 0.f;
    for (int k = k0; k < k1; ++k)
        acc += m[eid[k] * NC + c];
    node_in[idx] = acc;
}

// ---------------------------------------------------------------- BP: per-edge message update
__global__ void k_edge_update(const float* __restrict__ m, float* __restrict__ m_out,
                              const int* __restrict__ s0, const int* __restrict__ d0,
                              const float* __restrict__ rz, const float* __restrict__ phiu,
                              const float* __restrict__ phib,
                              const float* __restrict__ node_in) {
    __shared__ float pb[64];
    if (threadIdx.x < 64) pb[threadIdx.x] = phib[threadIdx.x];
    __syncthreads();
    int e = blockIdx.x * blockDim.x + threadIdx.x;
    if (e >= ED) return;
    int s, re, u;
    if (e < E_UND) { s = s0[e];          re = e + E_UND; u = e; }
    else           { s = d0[e - E_UND];  re = e - E_UND; u = e - E_UND; }
    float sc = rz[u];

    const float4 p0 = *reinterpret_cast<const float4*>(phiu + (size_t)s * NC);
    const float4 p1 = *reinterpret_cast<const float4*>(phiu + (size_t)s * NC + 4);
    const float4 n0 = *reinterpret_cast<const float4*>(node_in + (size_t)s * NC);
    const float4 n1 = *reinterpret_cast<const float4*>(node_in + (size_t)s * NC + 4);
    const float4 m0 = *reinterpret_cast<const float4*>(m + (size_t)re * NC);
    const float4 m1 = *reinterpret_cast<const float4*>(m + (size_t)re * NC + 4);

    float hm[NC];
    hm[0] = p0.x + n0.x - m0.x;  hm[1] = p0.y + n0.y - m0.y;
    hm[2] = p0.z + n0.z - m0.z;  hm[3] = p0.w + n0.w - m0.w;
    hm[4] = p1.x + n1.x - m1.x;  hm[5] = p1.y + n1.y - m1.y;
    hm[6] = p1.z + n1.z - m1.z;  hm[7] = p1.w + n1.w - m1.w;

    float out[NC];
#pragma unroll
    for (int j = 0; j < NC; ++j) {
        float t[NC];
        float mx = -1e30f;
#pragma unroll
        for (int i = 0; i < NC; ++i) {
            t[i] = hm[i] + sc * pb[i * 8 + j];
            mx = fmaxf(mx, t[i]);
        }
        float sum = 0.f;
#pragma unroll
        for (int i = 0; i < NC; ++i) sum += __expf(t[i] - mx);
        out[j] = mx + __logf(sum);
    }
    float mx2 = out[0];
#pragma unroll
    for (int j = 1; j < NC; ++j) mx2 = fmaxf(mx2, out[j]);
    float sum2 = 0.f;
#pragma unroll
    for (int j = 0; j < NC; ++j) sum2 += __expf(out[j] - mx2);
    float lse = mx2 + __logf(sum2);

    float4 o0, o1;
    o0.x = out[0] - lse; o0.y = out[1] - lse; o0.z = out[2] - lse; o0.w = out[3] - lse;
    o1.x = out[4] - lse; o1.y = out[5] - lse; o1.z = out[6] - lse; o1.w = out[7] - lse;
    *reinterpret_cast<float4*>(m_out + (size_t)e * NC)     = o0;
    *reinterpret_cast<float4*>(m_out + (size_t)e * NC + 4) = o1;
}

// ---------------------------------------------------------------- final log-softmax
__global__ void k_output(const float* __restrict__ phiu, const float* __restrict__ node_in,
                         float* __restrict__ out) {
    int v = blockIdx.x * blockDim.x + threadIdx.x;
    if (v >= N_NODES) return;
    float b[NC];
#pragma unroll
    for (int c = 0; c < NC; ++c) b[c] = phiu[v * NC + c] + node_in[v * NC + c];
    float mx = b[0];
#pragma unroll
    for (int c = 1; c < NC; ++c) mx = fmaxf(mx, b[c]);
    float sum = 0.f;
#pragma unroll
    for (int c = 0; c < NC; ++c) sum += __expf(b[c] - mx);
    float lse = mx + __logf(sum);
#pragma unroll
    for (int c = 0; c < NC; ++c) out[v * NC + c] = b[c] - lse;
}

// ---------------------------------------------------------------- launch
static inline int cdiv_i(long long a, long long b) { return (int)((a + b - 1) / b); }

extern "C" void kernel_launch(void* const* d_in, const int* in_sizes, int n_in,
                              void* d_out, int out_size, void* d_ws, size_t ws_size,
                              hipStream_t stream) {
    const float* x      = (const float*)d_in[0];
    const int*   ei     = (const int*)d_in[1];
    const unsigned char* mask = (const unsigned char*)d_in[2];
    const int*   target = (const int*)d_in[3];
    const float* W1     = (const float*)d_in[4];
    const float* b1     = (const float*)d_in[5];
    const float* W2     = (const float*)d_in[6];
    const float* b2     = (const float*)d_in[7];
    const float* binary = (const float*)d_in[8];
    const float* rz     = (const float*)d_in[9];
    const int* s0 = ei;
    const int* d0 = ei + E_UND;

    // ---- workspace layout (floats, then ints)
    float* ws   = (float*)d_ws;
    float* dinv = ws;                                   // 20480 (padded)
    float* H1   = dinv + 20480;                         // N*128
    float* h    = H1 + (long long)N_NODES * F_INN;      // N*128
    float* G2   = h  + (long long)N_NODES * F_INN;      // N*8
    float* phiu = G2 + N_NODES * NC;                    // N*8
    float* phib = phiu + N_NODES * NC;                  // 64
    float* mA   = phib + 64;                            // ED*8
    float* mB   = mA + (long long)ED * NC;              // ED*8
    float* node_in = mB + (long long)ED * NC;           // N*8
    int*   cnt    = (int*)(node_in + N_NODES * NC);     // 20480 (in-degree histogram)
    int*   ptr    = cnt + 20480;                        // 20480 (N+1 used)
    int*   cursor = ptr + 20480;                        // 20480
    int*   eid    = cursor + 20480;                     // ED
    int*   esrc   = eid + ED;                           // ED

    // ---- CSR of incoming directed edges + dinv
    k_zero_int<<<cdiv_i(N_NODES,256),256,0,stream>>>(cnt, N_NODES);
    k_indeg<<<cdiv_i(E_UND,256),256,0,stream>>>(s0, d0, cnt, E_UND);
    k_scan<<<1,1024,0,stream>>>(cnt, ptr, cursor);
    k_place<<<cdiv_i(ED,256),256,0,stream>>>(s0, d0, cursor, eid, esrc);
    k_dinv<<<cdiv_i(N_NODES,256),256,0,stream>>>(cnt, dinv, N_NODES);

    // ---- GCN layer 1 (WMMA GEMM + CSR gather, fused epilogue)
    k_gemm1<<<N_NODES/16, 32, 0, stream>>>(x, W1, H1);
    k_conv1<<<cdiv_i((long long)N_NODES*32,256),256,0,stream>>>(H1, ptr, esrc, dinv, b1, h);

    // ---- GCN layer 2 -> unary
    k_gemm2<<<N_NODES/16, 32, 0, stream>>>(h, W2, G2);
    k_conv2<<<cdiv_i(N_NODES*NC,256),256,0,stream>>>(G2, ptr, esrc, dinv, b2, phiu);

    // ---- potentials
    k_phi_u<<<cdiv_i(N_NODES,256),256,0,stream>>>(phiu, target, mask);
    k_phi_b<<<1,64,0,stream>>>(binary, phib);

    // ---- BP loop (double buffered messages, gather-based segment sums)
    k_fill<<<cdiv_i((long long)ED*NC,256),256,0,stream>>>(mA, 0.0f, ED*NC);
    float* cur = mA;
    float* nxt = mB;
    for (int it = 0; it < BP_ITERS; ++it) {
        k_gather_m<<<cdiv_i(N_NODES*NC,256),256,0,stream>>>(cur, ptr, eid, node_in);
        k_edge_update<<<cdiv_i(ED,256),256,0,stream>>>(cur, nxt, s0, d0, rz, phiu, phib, node_in);
        float* t = cur; cur = nxt; nxt = t;
    }

    // ---- beliefs + log_softmax
    k_gather_m<<<cdiv_i(N_NODES*NC,256),256,0,stream>>>(cur, ptr, eid, node_in);
    k_output<<<cdiv_i(N_NODES,256),256,0,stream>>>(phiu, node_in, (float*)d_out);
}